// GritTransformerLayer_76613626626560
// MI455X (gfx1250) — compile-verified
//
#include <hip/hip_runtime.h>

typedef __attribute__((ext_vector_type(2))) float v2f;
typedef __attribute__((ext_vector_type(4))) float f4;
typedef __attribute__((ext_vector_type(8))) float v8f;

constexpr int Nn = 100000;
constexpr int Ee = 800000;
constexpr int Dd = 64;
constexpr int Hh = 8;
constexpr int Dh = 8;
constexpr int BN_RPB = 2048;

// ---------------- helpers ----------------
__device__ __forceinline__ unsigned enc_f(float f) {
  unsigned u = __float_as_uint(f);
  return (u & 0x80000000u) ? ~u : (u | 0x80000000u);
}
__device__ __forceinline__ float dec_f(unsigned u) {
  return (u & 0x80000000u) ? __uint_as_float(u & 0x7FFFFFFFu) : __uint_as_float(~u);
}
__device__ __forceinline__ float signed_sqrt(float s) {
  return s > 0.f ? __builtin_sqrtf(s) : (s < 0.f ? -__builtin_sqrtf(-s) : 0.f);
}

__global__ void fill_u32_kernel(unsigned* __restrict__ p, unsigned v, int n) {
  int i = blockIdx.x * blockDim.x + threadIdx.x;
  if (i < n) p[i] = v;
}

// ---------------- WMMA f32 GEMM: one wave -> 16 x (NT*16) strip ----------------
// Out[M x Nc] = relu?(A[M x Kc] @ W[Kc x Nc] + bias) + resid, Nc = NT*16.
// A is loaded ONCE per k-step and reused across all NT accumulator tiles
// (cuts A-side VMEM traffic/instructions by NT vs one-tile-per-wave), and the
// NT independent V_WMMA_F32_16X16X4_F32 between same-accumulator reuses hide
// the WMMA->WMMA RAW latency (ISA 7.12.1).
template <int NT>
__global__ __launch_bounds__(256)
void wmma_gemm_kernel(const float* __restrict__ A, const float* __restrict__ W,
                      const float* __restrict__ bias, const float* __restrict__ resid,
                      float* __restrict__ Out, int M, int Kc, int relu) {
  constexpr int Nc = NT * 16;
  const int lane = threadIdx.x & 31;
  const int wave = blockIdx.x * (blockDim.x >> 5) + (threadIdx.x >> 5);
  if (wave >= (M >> 4)) return;            // wave-uniform: EXEC all-1s for WMMA
  const int m0 = wave << 4;

  // A 16x4 layout: lane&15 = M row; VGPR pair holds K = kofs, kofs+1
  // (kofs = 0 for lanes 0-15, 2 for lanes 16-31). B 4x16: lane&15 = N col.
  const int kofs = (lane >> 4) << 1;
  const int nlane = lane & 15;
  const float* arow = A + (size_t)(m0 + nlane) * Kc;

  v8f c[NT];
#pragma unroll
  for (int t = 0; t < NT; ++t) c[t] = v8f{0.f, 0.f, 0.f, 0.f, 0.f, 0.f, 0.f, 0.f};

  for (int k0 = 0; k0 < Kc; k0 += 4) {
    v2f a = *(const v2f*)(arow + k0 + kofs);
    const float* w0 = W + (size_t)(k0 + kofs) * Nc + nlane;
    const float* w1 = w0 + Nc;
#pragma unroll
    for (int t = 0; t < NT; ++t) {
      v2f b;
      b.x = w0[t * 16];
      b.y = w1[t * 16];
      c[t] = __builtin_amdgcn_wmma_f32_16x16x4_f32(false, a, false, b, (short)0, c[t],
                                                   false, false);
    }
  }

  // D layout: VGPR j -> row m0+j (lanes 0-15) / m0+8+j (lanes 16-31).
  const int mbase = m0 + ((lane >> 4) << 3);
  float bcol[NT];
#pragma unroll
  for (int t = 0; t < NT; ++t) bcol[t] = bias ? bias[t * 16 + nlane] : 0.f;
#pragma unroll
  for (int j = 0; j < 8; ++j) {
    const size_t rowo = (size_t)(mbase + j) * Nc + nlane;
#pragma unroll
    for (int t = 0; t < NT; ++t) {
      float v = c[t][j] + bcol[t];
      if (relu) v = v > 0.f ? v : 0.f;
      if (resid) v += resid[rowo + t * 16];
      Out[rowo + t * 16] = v;
    }
  }
}

// ---------------- edge kernels (b128 loads: operands are 32B-aligned) ----------------
__global__ void edge_score_kernel(const float* __restrict__ Qb, const float* __restrict__ Kb,
                                  const float* __restrict__ Ef, const float* __restrict__ Aw,
                                  const int* __restrict__ srcI, const int* __restrict__ dstI,
                                  float* __restrict__ S, float* __restrict__ score,
                                  unsigned* __restrict__ smaxU, float* __restrict__ deg) {
  int idx = blockIdx.x * blockDim.x + threadIdx.x;
  if (idx >= Ee * Hh) return;
  int e = idx >> 3, h = idx & 7;
  int src = srcI[e], dst = dstI[e];
  const f4* kk = (const f4*)(Kb + (size_t)src * Dd + h * Dh);
  const f4* qq = (const f4*)(Qb + (size_t)dst * Dd + h * Dh);
  const f4* ef = (const f4*)(Ef + (size_t)e * (2 * Dd) + h * (2 * Dh));  // [Ew(8)|Eb(8)]
  f4* so = (f4*)(S + (size_t)e * Dd + h * Dh);
  f4 kq[2] = {kk[0] + qq[0], kk[1] + qq[1]};
  f4 ew[2] = {ef[0], ef[1]};
  f4 eb[2] = {ef[2], ef[3]};
  float sc = 0.f;
#pragma unroll
  for (int p = 0; p < 2; ++p) {
    f4 ss;
#pragma unroll
    for (int i = 0; i < 4; ++i) {
      float t = kq[p][i] * ew[p][i];
      float v = signed_sqrt(t) + eb[p][i];
      v = v > 0.f ? v : 0.f;                 // relu -> e_t / wE
      ss[i] = v;
      sc += v * Aw[(p * 4 + i) * Hh + h];    // Aw is (Dh,H,1)
    }
    so[p] = ss;
  }
  sc = fminf(fmaxf(sc, -5.f), 5.f);
  score[idx] = sc;
  atomicMax(&smaxU[dst * Hh + h], enc_f(sc));
  if (h == 0) atomicAdd(&deg[dst], 1.f);
}

__global__ void edge_exp_kernel(const int* __restrict__ dstI,
                                const unsigned* __restrict__ smaxU,
                                float* __restrict__ score_alpha,  // in: score, out: exp(score-max)
                                float* __restrict__ ssum) {
  int idx = blockIdx.x * blockDim.x + threadIdx.x;
  if (idx >= Ee * Hh) return;
  int e = idx >> 3, h = idx & 7;
  int dst = dstI[e];
  float a = __expf(score_alpha[idx] - dec_f(smaxU[dst * Hh + h]));
  score_alpha[idx] = a;
  atomicAdd(&ssum[dst * Hh + h], a);
}

__global__ void edge_scatter_kernel(const int* __restrict__ srcI, const int* __restrict__ dstI,
                                    const float* __restrict__ Vb, const float* __restrict__ S,
                                    const float* __restrict__ alpha, const float* __restrict__ ssum,
                                    float* __restrict__ wV, float* __restrict__ rowV) {
  int idx = blockIdx.x * blockDim.x + threadIdx.x;
  if (idx >= Ee * Hh) return;
  int e = idx >> 3, h = idx & 7;
  int src = srcI[e], dst = dstI[e];
  float a = alpha[idx] / (ssum[dst * Hh + h] + 1e-16f);
  const f4* vv = (const f4*)(Vb + (size_t)src * Dd + h * Dh);
  const f4* se = (const f4*)(S + (size_t)e * Dd + h * Dh);
  float* wo = wV + (size_t)dst * Dd + h * Dh;
  float* ro = rowV + (size_t)dst * Dd + h * Dh;
  f4 v0 = vv[0], v1 = vv[1], s0 = se[0], s1 = se[1];
#pragma unroll
  for (int d = 0; d < 4; ++d) {
    atomicAdd(&wo[d], v0[d] * a);
    atomicAdd(&wo[d + 4], v1[d] * a);
    atomicAdd(&ro[d], s0[d] * a);
    atomicAdd(&ro[d + 4], s1[d] * a);
  }
}

// ---------------- node combine: VeRow einsum + degree scaling ----------------
__global__ void node_combine_kernel(const float* __restrict__ wV, const float* __restrict__ rowV,
                                    const float* __restrict__ VeRow, const float* __restrict__ deg,
                                    const float* __restrict__ dc, float* __restrict__ hout) {
  int idx = blockIdx.x * blockDim.x + threadIdx.x;
  if (idx >= Nn * Dd) return;
  int n = idx >> 6, c64 = idx & 63, h = c64 >> 3, cc = c64 & 7;
  const f4* rv = (const f4*)(rowV + (size_t)n * Dd + h * Dh);
  f4 r0 = rv[0], r1 = rv[1];
  float acc = wV[idx];
#pragma unroll
  for (int d = 0; d < 4; ++d) {
    acc += r0[d] * VeRow[(d * Hh + h) * Dh + cc];        // VeRow (Dh,H,Dh)
    acc += r1[d] * VeRow[((d + 4) * Hh + h) * Dh + cc];
  }
  float ld = __logf(deg[n] + 1.f);
  hout[idx] = acc * dc[c64 * 2] + acc * ld * dc[c64 * 2 + 1];  // deg_coef (1,D,2)
}

// ---------------- batch norm ----------------
__global__ __launch_bounds__(256)
void bn_stats_kernel(const float* __restrict__ X, int R, float* __restrict__ stats) {
  int col = threadIdx.x & 63;
  int sub = threadIdx.x >> 6;                // 4 sub-rows per block
  int r0 = blockIdx.x * BN_RPB + sub;
  int r1 = min(R, (blockIdx.x + 1) * BN_RPB);
  float s = 0.f, s2 = 0.f;
  for (int r = r0; r < r1; r += 4) {
    __builtin_prefetch(&X[(size_t)(r + 64) * 64 + col], 0, 1);
    float v = X[(size_t)r * 64 + col];
    s += v;
    s2 += v * v;
  }
  atomicAdd(&stats[col], s);
  atomicAdd(&stats[64 + col], s2);
}

__global__ void bn_apply_kernel(const float* __restrict__ X, int R,
                                const float* __restrict__ stats,
                                const float* __restrict__ g, const float* __restrict__ b,
                                float* __restrict__ out, int total) {
  int idx = blockIdx.x * blockDim.x + threadIdx.x;
  if (idx >= total) return;
  int col = idx & 63;
  float invR = 1.f / (float)R;
  float m = stats[col] * invR;
  float v = stats[64 + col] * invR - m * m;
  out[idx] = g[col] * (X[idx] - m) * rsqrtf(v + 1e-5f) + b[col];
}

// ---------------- host ----------------
extern "C" void kernel_launch(void* const* d_in, const int* in_sizes, int n_in,
                              void* d_out, int out_size, void* d_ws, size_t ws_size,
                              hipStream_t stream) {
  (void)in_sizes; (void)n_in; (void)out_size; (void)ws_size;
  const float* x         = (const float*)d_in[0];
  const float* edge_attr = (const float*)d_in[1];
  const int*   ei        = (const int*)d_in[2];
  const int*   srcI = ei;
  const int*   dstI = ei + Ee;
  const float* Wq = (const float*)d_in[3];  const float* bq = (const float*)d_in[4];
  const float* Wk = (const float*)d_in[5];  const float* bk = (const float*)d_in[6];
  const float* We = (const float*)d_in[7];  const float* be = (const float*)d_in[8];
  const float* Wv = (const float*)d_in[9];  const float* bv = (const float*)d_in[10];
  const float* Aw = (const float*)d_in[11]; const float* VeRow = (const float*)d_in[12];
  const float* Wo_h = (const float*)d_in[13]; const float* bo_h = (const float*)d_in[14];
  const float* Wo_e = (const float*)d_in[15]; const float* bo_e = (const float*)d_in[16];
  const float* dc  = (const float*)d_in[17];
  const float* g1h = (const float*)d_in[18]; const float* b1h = (const float*)d_in[19];
  const float* g1e = (const float*)d_in[20]; const float* b1e = (const float*)d_in[21];
  const float* Wf1 = (const float*)d_in[22]; const float* bf1 = (const float*)d_in[23];
  const float* Wf2 = (const float*)d_in[24]; const float* bf2 = (const float*)d_in[25];
  const float* g2h = (const float*)d_in[26]; const float* b2h = (const float*)d_in[27];

  float* ws = (float*)d_ws;
  size_t off = 0;
  auto take = [&](size_t n) { float* p = ws + off; off += n; return p; };
  float*    Qb    = take((size_t)Nn * Dd);
  float*    Kb    = take((size_t)Nn * Dd);
  float*    Vb    = take((size_t)Nn * Dd);
  float*    S     = take((size_t)Ee * Dd);       // e_t / wE
  float*    score = take((size_t)Ee * Hh);       // becomes alpha in-place
  float*    ssum  = take((size_t)Nn * Hh);
  unsigned* smaxU = (unsigned*)take((size_t)Nn * Hh);
  float*    wV    = take((size_t)Nn * Dd);
  float*    rowV  = take((size_t)Nn * Dd);
  float*    deg   = take((size_t)Nn);
  float*    hbuf  = take((size_t)Nn * Dd);
  float*    hres  = take((size_t)Nn * Dd);
  float*    hid   = take((size_t)Nn * 2 * Dd);
  float*    stats = take(3 * 128);
  float*    Ef    = take((size_t)Ee * 2 * Dd);   // dead after edge_score; reused for e1
  float*    e1    = Ef;

  float* out_h = (float*)d_out;
  float* out_e = out_h + (size_t)Nn * Dd;

  auto cdiv = [](long a, long b) { return (unsigned)((a + b - 1) / b); };

  // ---- init accumulators ----
  fill_u32_kernel<<<cdiv(Nn * Hh, 256), 256, 0, stream>>>(smaxU, 0x007FFFFFu, Nn * Hh); // enc(-inf)
  fill_u32_kernel<<<cdiv(Nn * Hh, 256), 256, 0, stream>>>((unsigned*)ssum, 0u, Nn * Hh);
  fill_u32_kernel<<<cdiv(Nn * Dd, 256), 256, 0, stream>>>((unsigned*)wV, 0u, Nn * Dd);
  fill_u32_kernel<<<cdiv(Nn * Dd, 256), 256, 0, stream>>>((unsigned*)rowV, 0u, Nn * Dd);
  fill_u32_kernel<<<cdiv(Nn, 256), 256, 0, stream>>>((unsigned*)deg, 0u, Nn);
  fill_u32_kernel<<<2, 256, 0, stream>>>((unsigned*)stats, 0u, 384);

  auto gemm = [&](const float* A, const float* Wm, const float* bias, const float* resid,
                  float* Out, int M, int Kc, int Nc, int relu) {
    unsigned grid = cdiv(M / 16, 8);   // one wave per 16-row strip, 8 waves/block
    if (Nc == 64)
      wmma_gemm_kernel<4><<<grid, 256, 0, stream>>>(A, Wm, bias, resid, Out, M, Kc, relu);
    else
      wmma_gemm_kernel<8><<<grid, 256, 0, stream>>>(A, Wm, bias, resid, Out, M, Kc, relu);
  };

  // ---- projections ----
  gemm(x, Wq, bq, nullptr, Qb, Nn, Dd, Dd, 0);
  gemm(x, Wk, bk, nullptr, Kb, Nn, Dd, Dd, 0);
  gemm(x, Wv, bv, nullptr, Vb, Nn, Dd, Dd, 0);
  gemm(edge_attr, We, be, nullptr, Ef, Ee, Dd, 2 * Dd, 0);

  // ---- edge attention ----
  edge_score_kernel<<<cdiv((long)Ee * Hh, 256), 256, 0, stream>>>(Qb, Kb, Ef, Aw, srcI, dstI,
                                                                  S, score, smaxU, deg);
  edge_exp_kernel<<<cdiv((long)Ee * Hh, 256), 256, 0, stream>>>(dstI, smaxU, score, ssum);
  edge_scatter_kernel<<<cdiv((long)Ee * Hh, 256), 256, 0, stream>>>(srcI, dstI, Vb, S, score,
                                                                    ssum, wV, rowV);
  node_combine_kernel<<<cdiv((long)Nn * Dd, 256), 256, 0, stream>>>(wV, rowV, VeRow, deg, dc, hbuf);

  // ---- output projections + residuals ----
  gemm(hbuf, Wo_h, bo_h, x, hres, Nn, Dd, Dd, 0);
  gemm(S, Wo_e, bo_e, edge_attr, e1, Ee, Dd, Dd, 0);

  // ---- BN1 (h and e) ----
  bn_stats_kernel<<<cdiv(Nn, BN_RPB), 256, 0, stream>>>(hres, Nn, stats);
  bn_apply_kernel<<<cdiv((long)Nn * Dd, 256), 256, 0, stream>>>(hres, Nn, stats, g1h, b1h,
                                                                hbuf, Nn * Dd);
  bn_stats_kernel<<<cdiv(Ee, BN_RPB), 256, 0, stream>>>(e1, Ee, stats + 256);
  bn_apply_kernel<<<cdiv((long)Ee * Dd, 256), 256, 0, stream>>>(e1, Ee, stats + 256, g1e, b1e,
                                                                out_e, Ee * Dd);

  // ---- FFN + BN2 ----
  gemm(hbuf, Wf1, bf1, nullptr, hid, Nn, Dd, 2 * Dd, 1);
  gemm(hid, Wf2, bf2, hbuf, hres, Nn, 2 * Dd, Dd, 0);
  bn_stats_kernel<<<cdiv(Nn, BN_RPB), 256, 0, stream>>>(hres, Nn, stats + 128);
  bn_apply_kernel<<<cdiv((long)Nn * Dd, 256), 256, 0, stream>>>(hres, Nn, stats + 128, g2h, b2h,
                                                                out_h, Nn * Dd);
}